// DiffGNNPlacement_29205777613568
// MI455X (gfx1250) — compile-verified
//
#include <hip/hip_runtime.h>
#include <math.h>

#define NN 100000      // nodes
#define NE 1600000     // edges
#define KIN 64
#define H1 64
#define H2 32

typedef __attribute__((ext_vector_type(2))) float v2f;
typedef __attribute__((ext_vector_type(8))) float v8f;

// ---------- small elementwise kernels ----------
__global__ void k_zero(float* p, int n) {
  int i = blockIdx.x * blockDim.x + threadIdx.x;
  if (i < n) p[i] = 0.0f;
}

// ew[e] = 0.5 + 1.0*sigmoid(logit); deg[col] += ew
__global__ void k_edge_ew_deg(const float* __restrict__ logits,
                              const int* __restrict__ ei,
                              float* __restrict__ ew,
                              float* __restrict__ deg) {
  int e = blockIdx.x * blockDim.x + threadIdx.x;
  if (e >= NE) return;
  float l = logits[e];
  float s = 0.5f + 1.0f / (1.0f + __expf(-l));
  ew[e] = s;
  int col = ei[NE + e];
  atomicAdd(&deg[col], s);
}

__global__ void k_dis(const float* __restrict__ deg, float* __restrict__ dis) {
  int i = blockIdx.x * blockDim.x + threadIdx.x;
  if (i < NN) {
    float d = deg[i] + 1.0f;           // self-loop weight 1
    dis[i] = rsqrtf(d);                // d > 0 always
  }
}

__global__ void k_norm(const int* __restrict__ ei,
                       const float* __restrict__ ew,
                       const float* __restrict__ dis,
                       float* __restrict__ norm) {
  int e = blockIdx.x * blockDim.x + threadIdx.x;
  if (e >= NE) return;
  int r = ei[e];
  int c = ei[NE + e];
  norm[e] = dis[r] * ew[e] * dis[c];
}

// ---------- WMMA f32 GEMM: H[M x NOUT] = X[M x K] * W[K x NOUT] ----------
// one wave = one 16x16 tile; blockDim.x = 32 * (NOUT/16); gridDim.x = M/16
template <int K, int NOUT>
__global__ void k_gemm_wmma(const float* __restrict__ X,
                            const float* __restrict__ W,
                            float* __restrict__ H) {
  const int wave = threadIdx.x >> 5;
  const int lane = threadIdx.x & 31;
  const int m0 = blockIdx.x * 16;
  const int n0 = wave * 16;
  const int half = lane >> 4;    // 0: K={k,k+1}, 1: K={k+2,k+3}
  const int l15 = lane & 15;

  // preload B fragments for all K/4 steps (W is tiny, L2-resident)
  v2f bfrag[K / 4];
#pragma unroll
  for (int kt = 0; kt < K / 4; ++kt) {
    int kk = kt * 4 + half * 2;
    int col = n0 + l15;
    v2f b;
    b.x = W[kk * NOUT + col];
    b.y = W[(kk + 1) * NOUT + col];
    bfrag[kt] = b;
  }

  const int row = m0 + l15;
  const float* xp = X + (size_t)row * K + half * 2;

  v8f acc = {};
#pragma unroll
  for (int kt = 0; kt < K / 4; ++kt) {
    v2f a = *(const v2f*)(xp + kt * 4);  // 8B-aligned contiguous pair
    acc = __builtin_amdgcn_wmma_f32_16x16x4_f32(
        /*neg_a=*/false, a, /*neg_b=*/false, bfrag[kt],
        /*c_mod=*/(short)0, acc, /*reuse_a=*/false, /*reuse_b=*/false);
  }

#pragma unroll
  for (int v = 0; v < 8; ++v) {
    int r = m0 + v + half * 8;
    H[(size_t)r * NOUT + n0 + l15] = acc[v];
  }
}

// out[i][c] = h[i][c] * dis[i]^2 + b[c]   (self-loop term + bias, full overwrite)
template <int NOUT>
__global__ void k_self(const float* __restrict__ h,
                       const float* __restrict__ dis,
                       const float* __restrict__ b,
                       float* __restrict__ out) {
  int idx = blockIdx.x * blockDim.x + threadIdx.x;
  if (idx >= NN * NOUT) return;
  int i = idx / NOUT;
  int c = idx - i * NOUT;
  float d = dis[i];
  out[idx] = h[idx] * (d * d) + b[c];
}

// out[col] += h[row] * norm[e]  ; one thread = one edge x 4 channels
template <int NOUT, int LOGG>
__global__ void k_scatter(const float* __restrict__ h,
                          const int* __restrict__ ei,
                          const float* __restrict__ norm,
                          float* __restrict__ out) {
  long long idx = (long long)blockIdx.x * blockDim.x + threadIdx.x;
  int e = (int)(idx >> LOGG);
  if (e >= NE) return;
  int g = (int)(idx & ((1 << LOGG) - 1));
  int row = ei[e];
  int col = ei[NE + e];
  float nw = norm[e];
  const float4 hv = *(const float4*)(h + (size_t)row * NOUT + g * 4);
  float* op = out + (size_t)col * NOUT + g * 4;
  atomicAdd(op + 0, hv.x * nw);
  atomicAdd(op + 1, hv.y * nw);
  atomicAdd(op + 2, hv.z * nw);
  atomicAdd(op + 3, hv.w * nw);
}

__global__ void k_relu(float* p, int n) {
  int i = blockIdx.x * blockDim.x + threadIdx.x;
  if (i < n) p[i] = fmaxf(p[i], 0.0f);
}

// logits = a2 @ lin_w + lin_b ; out[i] = {-logit, logit}
__global__ void k_final(const float* __restrict__ a2,
                        const float* __restrict__ lw,
                        const float* __restrict__ lb,
                        float* __restrict__ out) {
  int i = blockIdx.x * blockDim.x + threadIdx.x;
  if (i >= NN) return;
  const float* hp = a2 + (size_t)i * H2;
  float acc = lb[0];
#pragma unroll
  for (int c = 0; c < H2; ++c) acc = fmaf(hp[c], lw[c], acc);
  out[2 * i + 0] = -acc;
  out[2 * i + 1] = acc;
}

extern "C" void kernel_launch(void* const* d_in, const int* in_sizes, int n_in,
                              void* d_out, int out_size, void* d_ws, size_t ws_size,
                              hipStream_t stream) {
  const float* x      = (const float*)d_in[0];   // [NN, 64]
  const int*   ei     = (const int*)d_in[1];     // [2, NE] flat
  const float* elog   = (const float*)d_in[2];   // [NE]
  const float* W1     = (const float*)d_in[3];   // [64, 64]
  const float* b1     = (const float*)d_in[4];   // [64]
  const float* W2     = (const float*)d_in[5];   // [64, 32]
  const float* b2     = (const float*)d_in[6];   // [32]
  const float* lw     = (const float*)d_in[7];   // [32]
  const float* lb     = (const float*)d_in[8];   // [1]
  float* out = (float*)d_out;                    // [NN, 2]

  // workspace layout (floats)
  float* ws   = (float*)d_ws;
  float* deg  = ws;                         // NN
  float* dis  = deg + NN;                   // NN
  float* ew   = dis + NN;                   // NE
  float* nrm  = ew + NE;                    // NE
  float* h1   = nrm + NE;                   // NN*64
  float* a1   = h1 + (size_t)NN * H1;       // NN*64
  float* h2   = h1;                         // alias: h1 dead after layer-1 scatter
  float* a2   = h1 + (size_t)NN * H2;       // alias: second half of h1 region

  const int T = 256;
  // --- normalization terms (shared by both layers) ---
  k_zero<<<(NN + T - 1) / T, T, 0, stream>>>(deg, NN);
  k_edge_ew_deg<<<(NE + T - 1) / T, T, 0, stream>>>(elog, ei, ew, deg);
  k_dis<<<(NN + T - 1) / T, T, 0, stream>>>(deg, dis);
  k_norm<<<(NE + T - 1) / T, T, 0, stream>>>(ei, ew, dis, nrm);

  // --- layer 1: h1 = x@W1 ; a1 = scatter(h1,norm) + h1*dis^2 + b1 ; relu ---
  k_gemm_wmma<KIN, H1><<<NN / 16, 32 * (H1 / 16), 0, stream>>>(x, W1, h1);
  k_self<H1><<<(NN * H1 + T - 1) / T, T, 0, stream>>>(h1, dis, b1, a1);
  k_scatter<H1, 4><<<(int)(((long long)NE * (H1 / 4) + T - 1) / T), T, 0, stream>>>(h1, ei, nrm, a1);
  k_relu<<<(NN * H1 + T - 1) / T, T, 0, stream>>>(a1, NN * H1);

  // --- layer 2: h2 = a1@W2 ; a2 = scatter + self + b2 ; relu ---
  k_gemm_wmma<H1, H2><<<NN / 16, 32 * (H2 / 16), 0, stream>>>(a1, W2, h2);
  k_self<H2><<<(NN * H2 + T - 1) / T, T, 0, stream>>>(h2, dis, b2, a2);
  k_scatter<H2, 3><<<(int)(((long long)NE * (H2 / 4) + T - 1) / T), T, 0, stream>>>(h2, ei, nrm, a2);
  k_relu<<<(NN * H2 + T - 1) / T, T, 0, stream>>>(a2, NN * H2);

  // --- head ---
  k_final<<<(NN + T - 1) / T, T, 0, stream>>>(a2, lw, lb, out);
}